// AttenComm_62534723829943
// MI455X (gfx1250) — compile-verified
//
#include <hip/hip_runtime.h>
#include <hip/hip_bf16.h>

#define B_ 4
#define N_ 4
#define C_ 64
#define H_ 128
#define W_ 256
#define P_ (H_ * W_)
#define TP 32            // pixels per block (one 32-wide row segment)
#define FSTRIDE 258      // per-pixel LDS row stride (even -> 8B align; 2*pp banks distinct)

typedef __attribute__((ext_vector_type(2))) float v2f;
typedef __attribute__((ext_vector_type(8))) float v8f;

__global__ __launch_bounds__(256) void atten_comm_fused(
    const float* __restrict__ x,      // (B*N, C, H, W)
    const float* __restrict__ pair,   // (B, 5, 5, 4, 4)
    float* __restrict__ out)          // (B, C, H, W)
{
    __shared__ float f_s[TP * FSTRIDE];   // f_s[pp][n*64+c] warped features
    __shared__ float s_s[TP * N_];        // scores -> attn weights

    const int tid  = threadIdx.x;
    const int lane = tid & 31;
    const int b    = blockIdx.x >> 10;        // P_/TP = 1024 tiles per batch
    const int tile = blockIdx.x & 1023;
    const int p_base = tile * TP;

    const int pp = tid & (TP - 1);            // pixel within tile
    const int cg = tid >> 5;                  // channel group (8 channels each)
    const int p  = p_base + pp;
    const int h  = p >> 8;                    // W_ = 256
    const int w  = p & (W_ - 1);

    const float gx = fmaf((float)w, 2.0f / (float)(W_ - 1), -1.0f);
    const float gy = fmaf((float)h, 2.0f / (float)(H_ - 1), -1.0f);
    const float sx = 2.0f / (4.0f * 0.4f * (float)W_);   // DOWNSAMPLE_RATE * DISCRETE_RATIO
    const float sy = 2.0f / (4.0f * 0.4f * (float)H_);

    const float* xb = x + (size_t)b * N_ * C_ * P_;

    // ---------- Phase 1: bilinear affine warp of all agents into LDS ----------
    for (int n = 0; n < N_; ++n) {
        const float* Pm = pair + ((size_t)(b * 5 + 0) * 5 + n) * 16;
        float m00 = Pm[0];
        float m01 = Pm[1] * ((float)H_ / (float)W_);
        float m02 = Pm[3] * sx;
        float m10 = Pm[4] * ((float)W_ / (float)H_);
        float m11 = Pm[5];
        float m12 = Pm[7] * sy;

        float g0 = fmaf(m00, gx, fmaf(m01, gy, m02));
        float g1 = fmaf(m10, gx, fmaf(m11, gy, m12));
        float ix = (g0 + 1.0f) * 0.5f * (float)(W_ - 1);
        float iy = (g1 + 1.0f) * 0.5f * (float)(H_ - 1);

        float fx0 = floorf(ix), fy0 = floorf(iy);
        int x0 = (int)fx0, y0 = (int)fy0;
        int x1 = x0 + 1,   y1 = y0 + 1;
        float wx = ix - fx0, wy = iy - fy0;

        float vx0 = (x0 >= 0 && x0 < W_) ? 1.0f : 0.0f;
        float vx1 = (x1 >= 0 && x1 < W_) ? 1.0f : 0.0f;
        float vy0 = (y0 >= 0 && y0 < H_) ? 1.0f : 0.0f;
        float vy1 = (y1 >= 0 && y1 < H_) ? 1.0f : 0.0f;

        int x0c = min(max(x0, 0), W_ - 1);
        int x1c = min(max(x1, 0), W_ - 1);
        int y0c = min(max(y0, 0), H_ - 1);
        int y1c = min(max(y1, 0), H_ - 1);

        int o00 = y0c * W_ + x0c;
        int o01 = y0c * W_ + x1c;
        int o10 = y1c * W_ + x0c;
        int o11 = y1c * W_ + x1c;

        float k00 = (1.0f - wx) * (1.0f - wy) * vx0 * vy0;
        float k01 = wx * (1.0f - wy) * vx1 * vy0;
        float k10 = (1.0f - wx) * wy * vx0 * vy1;
        float k11 = wx * wy * vx1 * vy1;

        const float* xn = xb + (size_t)n * C_ * P_;
#pragma unroll
        for (int j = 0; j < 8; ++j) {
            int c = cg * 8 + j;
            const float* img = xn + (size_t)c * P_;
            float v = k00 * img[o00];
            v = fmaf(k01, img[o01], v);
            v = fmaf(k10, img[o10], v);
            v = fmaf(k11, img[o11], v);
            f_s[pp * FSTRIDE + n * C_ + c] = v;
        }
    }
    __syncthreads();

    // ---------- Phase 2: scores via chained V_WMMA_F32_16X16X4_F32 ----------
    // Wave wv handles (subtile st, agent m); S = Q0 * F_m^T, need only diag.
    {
        const int wv    = tid >> 5;      // 0..7
        const int st    = wv & 1;        // 16-pixel subtile
        const int m     = wv >> 1;       // agent 0..3
        const int row   = lane & 15;     // A: M index / B: N index (same pixel)
        const int khalf = lane >> 4;     // K sub-pair select
        const int abase = (st * 16 + row) * FSTRIDE;

        v8f acc = {};
#pragma unroll
        for (int k = 0; k < 16; ++k) {
            int coff = k * 4 + khalf * 2;
            v2f a, bb;
            a.x  = f_s[abase + 0  * C_ + coff];       // Q0[row][coff]
            a.y  = f_s[abase + 0  * C_ + coff + 1];   // Q0[row][coff+1]
            bb.x = f_s[abase + m * C_ + coff];        // F_m[row][coff] (B = F_m^T)
            bb.y = f_s[abase + m * C_ + coff + 1];
            acc = __builtin_amdgcn_wmma_f32_16x16x4_f32(
                false, a, false, bb, (short)0, acc, false, false);
        }
        // diag(i): i<8 -> acc[i] @ lane i ; i>=8 -> acc[i-8] @ lane 16+i (lane 24+r)
#pragma unroll
        for (int r = 0; r < 8; ++r) {
            float v = acc[r];
            if (lane == r)      s_s[(st * 16 + r)     * N_ + m] = v;
            if (lane == 24 + r) s_s[(st * 16 + 8 + r) * N_ + m] = v;
        }
    }
    __syncthreads();

    // ---------- Phase 3: per-pixel softmax over 4 agents (scale 1/sqrt(C)) ----------
    if (tid < TP) {
        float s0 = s_s[tid * N_ + 0] * 0.125f;
        float s1 = s_s[tid * N_ + 1] * 0.125f;
        float s2 = s_s[tid * N_ + 2] * 0.125f;
        float s3 = s_s[tid * N_ + 3] * 0.125f;
        float mx = fmaxf(fmaxf(s0, s1), fmaxf(s2, s3));
        float e0 = __expf(s0 - mx), e1 = __expf(s1 - mx);
        float e2 = __expf(s2 - mx), e3 = __expf(s3 - mx);
        float inv = 1.0f / (e0 + e1 + e2 + e3);
        s_s[tid * N_ + 0] = e0 * inv;
        s_s[tid * N_ + 1] = e1 * inv;
        s_s[tid * N_ + 2] = e2 * inv;
        s_s[tid * N_ + 3] = e3 * inv;
    }
    __syncthreads();

    // ---------- Phase 4: ctx = attn @ f ; coalesced store to (B,C,H,W) ----------
    {
        float a0 = s_s[pp * N_ + 0];
        float a1 = s_s[pp * N_ + 1];
        float a2 = s_s[pp * N_ + 2];
        float a3 = s_s[pp * N_ + 3];
        const float* fr = &f_s[pp * FSTRIDE];
#pragma unroll
        for (int j = 0; j < 8; ++j) {
            int c = cg * 8 + j;
            float ctx = a0 * fr[0 * C_ + c];
            ctx = fmaf(a1, fr[1 * C_ + c], ctx);
            ctx = fmaf(a2, fr[2 * C_ + c], ctx);
            ctx = fmaf(a3, fr[3 * C_ + c], ctx);
            out[((size_t)(b * C_ + c)) * P_ + p] = ctx;
        }
    }
}

extern "C" void kernel_launch(void* const* d_in, const int* in_sizes, int n_in,
                              void* d_out, int out_size, void* d_ws, size_t ws_size,
                              hipStream_t stream) {
    const float* x    = (const float*)d_in[0];   // (16, 64, 128, 256) f32
    const float* pair = (const float*)d_in[1];   // (4, 5, 5, 4, 4) f32
    // d_in[2] = record_len: unused by the reference computation
    float* out = (float*)d_out;                  // (4, 64, 128, 256) f32

    dim3 grid(B_ * (P_ / TP));   // 4096 blocks
    dim3 block(256);             // 8 wave32s
    atten_comm_fused<<<grid, block, 0, stream>>>(x, pair, out);
}